// GCNPredictor_39247411151092
// MI455X (gfx1250) — compile-verified
//
#include <hip/hip_runtime.h>

typedef float v2f __attribute__((ext_vector_type(2)));
typedef float v8f __attribute__((ext_vector_type(8)));

#define N_GRAPHS 2048

// ---------------- utility: zero fill ----------------
__global__ void zero_f32(float* __restrict__ p, long n) {
    long i = (long)blockIdx.x * blockDim.x + threadIdx.x;
    if (i < n) p[i] = 0.0f;
}

// ---------------- h0 = x @ W_emb : [N,35] x [35,8] ----------------
__global__ void embed_kernel(const float* __restrict__ x,
                             const float* __restrict__ W,
                             float* __restrict__ h0, int N) {
    int i = blockIdx.x * blockDim.x + threadIdx.x;
    if (i >= N) return;
    float acc[8];
#pragma unroll
    for (int j = 0; j < 8; ++j) acc[j] = 0.0f;
    const float* xr = x + (long)i * 35;
#pragma unroll
    for (int k = 0; k < 35; ++k) {
        float xv = xr[k];
#pragma unroll
        for (int j = 0; j < 8; ++j) acc[j] += xv * W[k * 8 + j];
    }
    float* o = h0 + (long)i * 8;
#pragma unroll
    for (int j = 0; j < 8; ++j) o[j] = acc[j];
}

// ---------------- agg8[dst] += h0[src], d=8 ----------------
__global__ void scatter_add8(const float* __restrict__ h,
                             const int* __restrict__ src,
                             const int* __restrict__ dst,
                             float* __restrict__ agg, long E) {
    long i = (long)blockIdx.x * blockDim.x + threadIdx.x;
    if (i >= E * 8) return;
    long e = i >> 3;
    int  c = (int)(i & 7);
    int  s = src[e], d = dst[e];
    atomicAdd(&agg[(long)d * 8 + c], h[(long)s * 8 + c]);
}

// ---------------- agg[dst] += h[src], d=128 (32 lanes x float4) ----------------
__global__ void scatter_add128(const float* __restrict__ h,
                               const int* __restrict__ src,
                               const int* __restrict__ dst,
                               float* __restrict__ agg, long E) {
    long i = (long)blockIdx.x * blockDim.x + threadIdx.x;
    if (i >= E * 32) return;
    long e = i >> 5;
    int  l = (int)(i & 31);
    int  s = src[e], d = dst[e];
    const float4 v = *(const float4*)&h[(long)s * 128 + l * 4];
    float* o = &agg[(long)d * 128 + l * 4];
    atomicAdd(o + 0, v.x);
    atomicAdd(o + 1, v.y);
    atomicAdd(o + 2, v.z);
    atomicAdd(o + 3, v.w);
}

// ---------------- H = relu(agg8 @ W_g0) : [N,8] x [8,128], WMMA f32 16x16x4 ----------------
// one wave (32 threads) per 16x16 output tile; K=8 -> 2 wmma steps
__global__ __launch_bounds__(32) void gemm8_relu(const float* __restrict__ A,
                                                 const float* __restrict__ W,
                                                 float* __restrict__ H, int N) {
    int lane = threadIdx.x;
    int m0 = blockIdx.x * 16;
    int n0 = blockIdx.y * 16;
    int mrow = m0 + (lane & 15);
    int mm = mrow < N ? mrow : N - 1;   // clamp loads; stores handled below
    int kb = 2 * (lane >> 4);           // K sub-pair owned by this half-wave
    int nn = n0 + (lane & 15);

    v8f c = {};
#pragma unroll
    for (int s = 0; s < 2; ++s) {
        int k0 = s * 4 + kb;
        v2f a = *(const v2f*)&A[(long)mm * 8 + k0];     // A[m][k0], A[m][k0+1]
        v2f b;
        b.x = W[(long)k0 * 128 + nn];                   // B[k0][n]
        b.y = W[(long)(k0 + 1) * 128 + nn];             // B[k0+1][n]
        c = __builtin_amdgcn_wmma_f32_16x16x4_f32(false, a, false, b,
                                                  (short)0, c, false, false);
    }
    int col = n0 + (lane & 15);
    int rbase = m0 + 8 * (lane >> 4);
    if (m0 + 16 <= N) {                  // uniform fast path: full tile
#pragma unroll
        for (int r = 0; r < 8; ++r)
            H[(long)(rbase + r) * 128 + col] = fmaxf(c[r], 0.0f);
    } else {                             // remainder tile
#pragma unroll
        for (int r = 0; r < 8; ++r) {
            int row = rbase + r;
            if (row < N) H[(long)row * 128 + col] = fmaxf(c[r], 0.0f);
        }
    }
}

// ---------------- H = relu(agg @ Wg) + H (in place) : [N,128] x [128,128] ----------------
// block = 128 threads = 4 waves; block tile = 64(M) x 32(N); each wave owns a
// 16x32 sub-tile (two v8f accumulators -> 2 wmma per K-step, A load amortized).
// A tile (64x128, 32KB) and transposed W tile (32 cols x 128 k, 16KB) in LDS.
__global__ __launch_bounds__(128) void gemm128_relu_res(const float* __restrict__ Agg,
                                                        const float* __restrict__ W,
                                                        float* __restrict__ H, int N) {
    __shared__ float ldsA[64 * 128];   // 32 KB
    __shared__ float ldsB[32 * 128];   // 16 KB, ldsB[n*128 + k] = W[k][n0+n]

    int tid  = threadIdx.x;            // 0..127
    int wave = tid >> 5;
    int lane = tid & 31;
    int m0 = blockIdx.x * 64;
    int n0 = blockIdx.y * 32;

    // stage A: 64 rows x 128 floats, coalesced float4 (2048 float4 / 128 threads)
#pragma unroll
    for (int it = 0; it < 16; ++it) {
        int idx  = it * 128 + tid;     // float4 index within tile
        int row  = idx >> 5;           // 32 float4 per row
        int c4   = idx & 31;
        int grow = m0 + row;
        int gr   = grow < N ? grow : N - 1;
        float4 v = *(const float4*)&Agg[(long)gr * 128 + c4 * 4];
        *(float4*)&ldsA[row * 128 + c4 * 4] = v;
    }
    // stage W tile transposed: thread tid owns k = tid, 32 columns
    {
        int k = tid;
#pragma unroll
        for (int n = 0; n < 32; ++n)
            ldsB[n * 128 + k] = W[(long)k * 128 + n0 + n];
    }
    __syncthreads();

    int mloc = wave * 16 + (lane & 15);
    int kb   = 2 * (lane >> 4);
    int nn   = lane & 15;

    v8f c0 = {};
    v8f c1 = {};
#pragma unroll
    for (int s = 0; s < 32; ++s) {
        int k0 = s * 4 + kb;
        v2f a  = *(const v2f*)&ldsA[mloc * 128 + k0];        // ds_load_b64
        v2f b0 = *(const v2f*)&ldsB[nn * 128 + k0];
        v2f b1 = *(const v2f*)&ldsB[(nn + 16) * 128 + k0];
        c0 = __builtin_amdgcn_wmma_f32_16x16x4_f32(false, a, false, b0,
                                                   (short)0, c0, false, false);
        c1 = __builtin_amdgcn_wmma_f32_16x16x4_f32(false, a, false, b1,
                                                   (short)0, c1, false, false);
    }

    // relu + residual, in place (each element owned by exactly one lane)
    int col0  = n0 + nn;
    int col1  = n0 + 16 + nn;
    int rbase = m0 + wave * 16 + 8 * (lane >> 4);
    if (m0 + 64 <= N) {                  // uniform fast path: full block tile
#pragma unroll
        for (int r = 0; r < 8; ++r) {
            long o0 = (long)(rbase + r) * 128 + col0;
            long o1 = (long)(rbase + r) * 128 + col1;
            H[o0] = fmaxf(c0[r], 0.0f) + H[o0];
            H[o1] = fmaxf(c1[r], 0.0f) + H[o1];
        }
    } else {                             // remainder tile
#pragma unroll
        for (int r = 0; r < 8; ++r) {
            int row = rbase + r;
            if (row < N) {
                long o0 = (long)row * 128 + col0;
                long o1 = (long)row * 128 + col1;
                H[o0] = fmaxf(c0[r], 0.0f) + H[o0];
                H[o1] = fmaxf(c1[r], 0.0f) + H[o1];
            }
        }
    }
}

// ---------------- g[graph_ids[node]] += H[node], d=128 ----------------
__global__ void pool_kernel(const float* __restrict__ H,
                            const int* __restrict__ gid,
                            float* __restrict__ g, int N) {
    long i = (long)blockIdx.x * blockDim.x + threadIdx.x;
    if (i >= (long)N * 128) return;
    int node = (int)(i >> 7);
    int c    = (int)(i & 127);
    atomicAdd(&g[(long)gid[node] * 128 + c], H[i]);
}

// ---------------- pred = relu(g @ W_p1) @ W_p2 + b ----------------
__global__ void head_kernel(const float* __restrict__ g,
                            const float* __restrict__ W1,
                            const float* __restrict__ W2,
                            const float* __restrict__ b2,
                            float* __restrict__ out) {
    int gi = blockIdx.x * blockDim.x + threadIdx.x;
    if (gi >= N_GRAPHS) return;
    const float* gr = g + (long)gi * 128;
    float acc = b2[0];
#pragma unroll 4
    for (int j = 0; j < 64; ++j) {
        float t = 0.0f;
        for (int k = 0; k < 128; ++k) t += gr[k] * W1[k * 64 + j];
        acc += fmaxf(t, 0.0f) * W2[j];
    }
    out[gi] = acc;
}

extern "C" void kernel_launch(void* const* d_in, const int* in_sizes, int n_in,
                              void* d_out, int out_size, void* d_ws, size_t ws_size,
                              hipStream_t stream) {
    const float* x     = (const float*)d_in[0];
    const float* W_emb = (const float*)d_in[1];
    const float* W_g0  = (const float*)d_in[2];
    const float* W_g1  = (const float*)d_in[3];
    const float* W_g2  = (const float*)d_in[4];
    const float* W_p1  = (const float*)d_in[5];
    const float* W_p2  = (const float*)d_in[6];
    const float* b_p2  = (const float*)d_in[7];
    const int*   esrc  = (const int*)d_in[8];
    const int*   edst  = (const int*)d_in[9];
    const int*   gids  = (const int*)d_in[10];

    int  N = in_sizes[0] / 35;
    long E = in_sizes[8];
    float* pred = (float*)d_out;

    // workspace carve-up (floats): ~110 MB total
    float* ws   = (float*)d_ws;
    float* h0   = ws;                          // N*8
    float* agg8 = h0   + (size_t)N * 8;        // N*8
    float* H    = agg8 + (size_t)N * 8;        // N*128
    float* AGG  = H    + (size_t)N * 128;      // N*128
    float* G    = AGG  + (size_t)N * 128;      // 2048*128

    const int T = 256;
    long n8   = (long)N * 8;
    long n128 = (long)N * 128;
    long nG   = (long)N_GRAPHS * 128;

    dim3 gtile16((N + 15) / 16, 8);   // 16x16 tiles over [N,128]
    dim3 gtile64((N + 63) / 64, 4);   // 64x32 tiles over [N,128]

    // embed
    embed_kernel<<<(N + T - 1) / T, T, 0, stream>>>(x, W_emb, h0, N);

    // layer 0: agg8 = scatter(h0); H = relu(agg8 @ W_g0)
    zero_f32<<<(int)((n8 + T - 1) / T), T, 0, stream>>>(agg8, n8);
    scatter_add8<<<(int)((E * 8 + T - 1) / T), T, 0, stream>>>(h0, esrc, edst, agg8, E);
    gemm8_relu<<<gtile16, 32, 0, stream>>>(agg8, W_g0, H, N);

    // layer 1: AGG = scatter(H); H = relu(AGG @ W_g1) + H
    zero_f32<<<(int)((n128 + T - 1) / T), T, 0, stream>>>(AGG, n128);
    scatter_add128<<<(int)((E * 32 + T - 1) / T), T, 0, stream>>>(H, esrc, edst, AGG, E);
    gemm128_relu_res<<<gtile64, 128, 0, stream>>>(AGG, W_g1, H, N);

    // layer 2
    zero_f32<<<(int)((n128 + T - 1) / T), T, 0, stream>>>(AGG, n128);
    scatter_add128<<<(int)((E * 32 + T - 1) / T), T, 0, stream>>>(H, esrc, edst, AGG, E);
    gemm128_relu_res<<<gtile64, 128, 0, stream>>>(AGG, W_g2, H, N);

    // pooling + head
    zero_f32<<<(int)((nG + T - 1) / T), T, 0, stream>>>(G, nG);
    pool_kernel<<<(int)((n128 + T - 1) / T), T, 0, stream>>>(H, gids, G, N);
    head_kernel<<<(N_GRAPHS + T - 1) / T, T, 0, stream>>>(G, W_p1, W_p2, b_p2, pred);
}